// Global_Attention_61744449847590
// MI455X (gfx1250) — compile-verified
//
#include <hip/hip_runtime.h>
#include <hip/hip_bf16.h>

typedef __attribute__((ext_vector_type(2))) float v2f;
typedef __attribute__((ext_vector_type(8))) float v8f;

#define B_  16
#define N_  96
#define H_  256
#define BIN_ 11

// ---------------------------------------------------------------------------
// One 16x16 fp32 output tile per wave: C(16x16) = A(16xK) * B(Kx16),
// accumulated with V_WMMA_F32_16X16X4_F32 (K-step = 4).
// A row-major with leading dim lda, B row-major with leading dim ldb.
// ISA layouts (CDNA5 ISA):
//   A 16x4 : VGPR0 = K0 (lanes0-15) / K2 (lanes16-31); VGPR1 = K1 / K3
//   B 4x16 : lanes0-15 = K-half {0,1}, lanes16-31 = K-half {2,3}; col = lane&15
//   D      : VGPR r -> row (r + 8*(lane>>4)), col = lane&15
// ---------------------------------------------------------------------------
__device__ inline v8f wmma_f32_tile(const float* __restrict__ A, int lda,
                                    const float* __restrict__ B, int ldb,
                                    int K, int m0, int n0, int lane) {
    const int half = lane >> 4;
    const int lrow = lane & 15;
    v8f c = {0.f, 0.f, 0.f, 0.f, 0.f, 0.f, 0.f, 0.f};
    const float* __restrict__ arow = A + (size_t)(m0 + lrow) * lda;
    const float* __restrict__ bcol = B + n0 + lrow;
    for (int kk = 0; kk < K; kk += 4) {
        const int ka = kk + 2 * half;
        v2f a, b;
        a.x = arow[ka];
        a.y = arow[ka + 1];
        b.x = bcol[(size_t)ka * ldb];
        b.y = bcol[(size_t)(ka + 1) * ldb];
        c = __builtin_amdgcn_wmma_f32_16x16x4_f32(false, a, false, b,
                                                  (short)0, c, false, false);
    }
    return c;
}

// ---------------------------------------------------------------------------
// Kernel 1: P[b,n,k] = sum_h inputs[b,n,h] * W_atom[h,k]
// Per batch: M=96, N=256, K=256 -> 6x16 = 96 tiles; 16 batches -> 1536 tiles.
// blockDim = (32,4): 4 waves per block, one tile per wave.
// ---------------------------------------------------------------------------
__global__ void __launch_bounds__(128)
p_gemm_kernel(const float* __restrict__ in, const float* __restrict__ Wa,
              float* __restrict__ P) {
    const int lane = threadIdx.x;
    const int tile = blockIdx.x * 4 + threadIdx.y;      // 0..1535
    const int b  = tile / 96;
    const int t  = tile % 96;
    const int m0 = (t / 16) * 16;
    const int n0 = (t % 16) * 16;

    const float* A = in + (size_t)b * N_ * H_;
    v8f c = wmma_f32_tile(A, H_, Wa, H_, H_, m0, n0, lane);

    float* Pb = P + (size_t)b * N_ * H_;
    const int half = lane >> 4, lrow = lane & 15;
#pragma unroll
    for (int r = 0; r < 8; ++r)
        Pb[(size_t)(m0 + r + 8 * half) * H_ + n0 + lrow] = c[r];
}

// ---------------------------------------------------------------------------
// Kernel 2: one wave per (b,i,j) pair.
//   score[b,i,j] = sigmoid( sum_k relu(P_i[k]+P_j[k]+bin.W_bin[:,k]+b_bin[k])
//                           * w_score[k] + b_score )
//   atom_pair[b,i,j,:] = inputs[b,i,:] + inputs[b,j,:]   (streamed float4)
// ---------------------------------------------------------------------------
__global__ void __launch_bounds__(256)
score_pair_kernel(const float* __restrict__ in,
                  const float* __restrict__ binf,
                  const float* __restrict__ Wb,
                  const float* __restrict__ bbin,
                  const float* __restrict__ wsc,
                  const float* __restrict__ bsc,
                  const float* __restrict__ P,
                  float* __restrict__ score,
                  float* __restrict__ atom_pair) {
    const int lane = threadIdx.x & 31;
    const int wid  = (blockIdx.x * blockDim.x + threadIdx.x) >> 5; // pair id
    const int b  = wid / (N_ * N_);
    const int ij = wid % (N_ * N_);
    const int i  = ij / N_;
    const int j  = ij % N_;

    // broadcast the 11 bin features across the wave (1 load + shuffles)
    float bfl = (lane < BIN_) ? binf[(size_t)wid * BIN_ + lane] : 0.f;
    float bf[BIN_];
#pragma unroll
    for (int c = 0; c < BIN_; ++c) bf[c] = __shfl(bfl, c, 32);

    const float* __restrict__ Pi = P + (size_t)(b * N_ + i) * H_;
    const float* __restrict__ Pj = P + (size_t)(b * N_ + j) * H_;

    float acc = 0.f;
#pragma unroll
    for (int t = 0; t < H_ / 32; ++t) {
        const int k = lane + 32 * t;
        float q = bbin[k];
#pragma unroll
        for (int c = 0; c < BIN_; ++c) q = fmaf(bf[c], Wb[c * H_ + k], q);
        float h = Pi[k] + Pj[k] + q;
        h = fmaxf(h, 0.f);
        acc = fmaf(h, wsc[k], acc);
    }
#pragma unroll
    for (int off = 16; off > 0; off >>= 1) acc += __shfl_xor(acc, off, 32);

    if (lane == 0) {
        const float s = 1.f / (1.f + __expf(-(acc + bsc[0])));
        score[wid] = s;
    }

    // fused atom_pair write: 256 floats = 64 float4, 2 per lane (coalesced)
    const float4* __restrict__ Ii = (const float4*)(in + (size_t)(b * N_ + i) * H_);
    const float4* __restrict__ Ij = (const float4*)(in + (size_t)(b * N_ + j) * H_);
    float4* __restrict__ AP = (float4*)(atom_pair + (size_t)wid * H_);
#pragma unroll
    for (int t = 0; t < 2; ++t) {
        const int x = lane + 32 * t;
        const float4 u = Ii[x];
        const float4 v = Ij[x];
        AP[x] = make_float4(u.x + v.x, u.y + v.y, u.z + v.z, u.w + v.w);
    }
}

// ---------------------------------------------------------------------------
// Kernel 3: context[b] (96x256) = score[b] (96x96) @ inputs[b] (96x256)
// Same WMMA tile routine, K=96 (24 wmma steps per tile).
// ---------------------------------------------------------------------------
__global__ void __launch_bounds__(128)
ctx_gemm_kernel(const float* __restrict__ score, const float* __restrict__ in,
                float* __restrict__ ctx) {
    const int lane = threadIdx.x;
    const int tile = blockIdx.x * 4 + threadIdx.y;      // 0..1535
    const int b  = tile / 96;
    const int t  = tile % 96;
    const int m0 = (t / 16) * 16;
    const int n0 = (t % 16) * 16;

    v8f c = wmma_f32_tile(score + (size_t)b * N_ * N_, N_,
                          in + (size_t)b * N_ * H_, H_, N_, m0, n0, lane);

    float* C = ctx + (size_t)b * N_ * H_;
    const int half = lane >> 4, lrow = lane & 15;
#pragma unroll
    for (int r = 0; r < 8; ++r)
        C[(size_t)(m0 + r + 8 * half) * H_ + n0 + lrow] = c[r];
}

// ---------------------------------------------------------------------------
extern "C" void kernel_launch(void* const* d_in, const int* in_sizes, int n_in,
                              void* d_out, int out_size, void* d_ws, size_t ws_size,
                              hipStream_t stream) {
    const float* in   = (const float*)d_in[0];   // (B,N,H)
    const float* binf = (const float*)d_in[1];   // (B,N,N,BIN)
    const float* Wa   = (const float*)d_in[2];   // (H,H)
    const float* Wb   = (const float*)d_in[3];   // (BIN,H)
    const float* bbin = (const float*)d_in[4];   // (H,)
    const float* wsc  = (const float*)d_in[5];   // (H,1)
    const float* bsc  = (const float*)d_in[6];   // (1,)

    float* ctx       = (float*)d_out;                            // (B,N,H)
    float* atom_pair = (float*)d_out + (size_t)B_ * N_ * H_;     // (B,N,N,H)

    float* P     = (float*)d_ws;                                 // B*N*H floats
    float* score = (float*)d_ws + (size_t)B_ * N_ * H_;          // B*N*N floats

    // Kernel 1: 1536 tiles, 4 waves/block
    p_gemm_kernel<<<dim3(1536 / 4), dim3(32, 4), 0, stream>>>(in, Wa, P);

    // Kernel 2: 16*96*96 = 147456 waves, 8 waves/block
    const int pairs = B_ * N_ * N_;
    score_pair_kernel<<<dim3(pairs / 8), dim3(256), 0, stream>>>(
        in, binf, Wb, bbin, wsc, bsc, P, score, atom_pair);

    // Kernel 3: 1536 tiles, 4 waves/block
    ctx_gemm_kernel<<<dim3(1536 / 4), dim3(32, 4), 0, stream>>>(score, in, ctx);
}